// WalkLayer_25142738550817
// MI455X (gfx1250) — compile-verified
//
#include <hip/hip_runtime.h>

// WalkLayer on MI455X (gfx1250, wave32).
// Shapes fixed by the reference: B=2, N=64, F=128, ITERS=2, BETA=0.9.

typedef float v2f __attribute__((ext_vector_type(2)));
typedef float v8f __attribute__((ext_vector_type(8)));

#define B_    2
#define N_    64
#define F_    128
#define BETA_ 0.9f
#define M_ROWS (B_ * N_ * N_)   // 8192 rows for the (b,i,j)-flattened GEMM

// ---------------------------------------------------------------------------
// Kernel 0: g0 = adj ? graph : 0      (vectorized float4, 32 vec4 per (b,i,j))
// ---------------------------------------------------------------------------
__global__ __launch_bounds__(256)
void init_graph_k(const float4* __restrict__ graph,
                  const int* __restrict__ adj,
                  float4* __restrict__ g0) {
  const int t = blockIdx.x * blockDim.x + threadIdx.x;   // 0 .. B*N*N*F/4-1
  const int bij = t >> 5;                                // t / (F_/4)
  float4 v = graph[t];
  if (adj[bij] == 0) v = make_float4(0.f, 0.f, 0.f, 0.f);
  g0[t] = v;
}

// ---------------------------------------------------------------------------
// Kernel 1: gW[r, :] = g[r, :] @ W    (M=8192, K=128, N=128), f32 WMMA.
// One wave per 16x16 D tile; block of 8 waves covers one 16-row stripe
// across all 8 column tiles (A rows re-read from L1/L2).
//
// V_WMMA_F32_16X16X4_F32 fragment layout (ISA 7.12.2):
//   A (16x4):  lane<16 -> M=lane,  regs {K=k0+0, k0+1}; lane>=16 -> {k0+2, k0+3}
//   B (4x16):  lane%16 = N;        regs {K=k0+0, k0+1} / {k0+2, k0+3} by lane/16
//   C/D:       reg r -> row (r + 8*(lane/16)), col = lane%16
// ---------------------------------------------------------------------------
__global__ __launch_bounds__(256)
void gemm_gW_wmma(const float* __restrict__ A,   // [8192,128] row-major
                  const float* __restrict__ W,   // [128,128]  row-major (k,l)
                  float* __restrict__ D) {       // [8192,128]
  const int lane  = threadIdx.x & 31;
  const int wave  = threadIdx.x >> 5;     // 0..7  -> column tile
  const int mbase = blockIdx.x * 16;      // row tile base
  const int nbase = wave * 16;
  const int lo    = lane & 15;            // M for A-frag, N for B/D-frag
  const int hi    = lane >> 4;            // selects K-pair / row-half

  v8f c = {};
  const float* arow = A + (size_t)(mbase + lo) * F_;

#pragma unroll
  for (int kk = 0; kk < F_ / 4; ++kk) {
    const int k0 = kk * 4 + 2 * hi;
    v2f a, b;
    a.x = arow[k0 + 0];                     // contiguous pair -> b64 load
    a.y = arow[k0 + 1];
    b.x = W[(size_t)(k0 + 0) * F_ + nbase + lo];
    b.y = W[(size_t)(k0 + 1) * F_ + nbase + lo];
    // D = A*B + C  (f32 in, f32 accum -- exact precision match to reference)
    c = __builtin_amdgcn_wmma_f32_16x16x4_f32(false, a, false, b,
                                              (short)0, c, false, false);
  }

#pragma unroll
  for (int r = 0; r < 8; ++r) {
    D[(size_t)(mbase + r + 8 * hi) * F_ + nbase + lo] = c[r];
  }
}

// ---------------------------------------------------------------------------
// Kernel 2: one wave32 per (b,i,m); lane owns 4 k-values.
//   ng[k] = sum_{j active} sigmoid(gW[b,i,j,k] * old[b,j,m,k])
//   active(j): j!=m, j!=i, i!=m, mask[b,i,j,m]!=0, and not all-k products==0
//   g_new = active_any ? ng + beta*(old - ng) : old     (beta_mat k-uniform)
// All skip branches are wave-uniform => EXEC stays full; the all-zero-over-k
// test is a single in-wave __all (wave32 ballot), no LDS traffic.
// ---------------------------------------------------------------------------
__global__ __launch_bounds__(256)
void walk_step(const float* __restrict__ g_old,  // [B,N,N,F]
               const float* __restrict__ gW,     // [B,N,N,F]
               const int* __restrict__ mask,     // [B,N,N,N]
               float* __restrict__ g_new) {      // [B,N,N,F]
  const int lane = threadIdx.x & 31;
  const int wid  = blockIdx.x * (blockDim.x >> 5) + (threadIdx.x >> 5); // (b,i,m)
  const int m  = wid & (N_ - 1);
  const int bi = wid >> 6;            // b*N + i
  const int i  = bi & (N_ - 1);
  const int b  = bi >> 6;
  const int k0 = lane * 4;

  const float4 old_im = *(const float4*)(g_old + (size_t)wid * F_ + k0);
  float4* outp = (float4*)(g_new + (size_t)wid * F_ + k0);

  if (i == m) {                       // every j masked -> row passes through
    *outp = old_im;
    return;
  }

  float a0 = 0.f, a1 = 0.f, a2 = 0.f, a3 = 0.f;
  bool any_active = false;
  const int bN = b * N_;

  for (int j = 0; j < N_; ++j) {
    if (j == m || j == i) continue;                       // diag masks (uniform)
    if (mask[((size_t)bi * N_ + j) * N_ + m] == 0) continue;  // uniform scalar

    const float4 gw = *(const float4*)(gW    + ((size_t)bi * N_ + j) * F_ + k0);
    const float4 go = *(const float4*)(g_old + ((size_t)(bN + j) * N_ + m) * F_ + k0);
    const float p0 = gw.x * go.x, p1 = gw.y * go.y;
    const float p2 = gw.z * go.z, p3 = gw.w * go.w;

    const int lane_zero = (p0 == 0.f) & (p1 == 0.f) & (p2 == 0.f) & (p3 == 0.f);
    if (__all(lane_zero)) continue;   // whole k-vector zero -> -inf vector

    any_active = true;
    a0 += 1.f / (1.f + __expf(-p0));
    a1 += 1.f / (1.f + __expf(-p1));
    a2 += 1.f / (1.f + __expf(-p2));
    a3 += 1.f / (1.f + __expf(-p3));
  }

  float4 r;
  if (any_active) {
    r.x = a0 + BETA_ * (old_im.x - a0);
    r.y = a1 + BETA_ * (old_im.y - a1);
    r.z = a2 + BETA_ * (old_im.z - a2);
    r.w = a3 + BETA_ * (old_im.w - a3);
  } else {
    r = old_im;                       // beta_mat == 1.0 and ng == 0
  }
  *outp = r;
}

// ---------------------------------------------------------------------------
// Launch: init -> (gemm, walk) x 2.  Ping-pong via workspace:
//   bufA (4MB) = g0, bufGW (4MB) = gW scratch, bufB (4MB) = iter-1 output.
// Final iteration writes d_out. Total ws use: 12 MB.
// ---------------------------------------------------------------------------
extern "C" void kernel_launch(void* const* d_in, const int* in_sizes, int n_in,
                              void* d_out, int out_size, void* d_ws, size_t ws_size,
                              hipStream_t stream) {
  const float* graph = (const float*)d_in[0];   // [B,N,N,F] f32
  const int*   adj   = (const int*)d_in[1];     // [B,N,N]
  const int*   mask  = (const int*)d_in[2];     // [B,N,N,N]
  const float* W     = (const float*)d_in[3];   // [F,F] f32
  float*       out   = (float*)d_out;           // [B,N,N,F] f32

  const size_t gelems = (size_t)M_ROWS * F_;    // 1,048,576
  float* bufA  = (float*)d_ws;
  float* bufGW = bufA + gelems;
  float* bufB  = bufGW + gelems;

  const int initBlocks = (int)(gelems / 4 / 256);   // 1024
  const int gemmBlocks = M_ROWS / 16;               // 512 (8 waves/block)
  const int walkBlocks = M_ROWS / 8;                // 1024 (8 waves/block)

  // g0 = where(adj, graph, 0)
  init_graph_k<<<initBlocks, 256, 0, stream>>>((const float4*)graph, adj,
                                               (float4*)bufA);
  // iteration 1
  gemm_gW_wmma<<<gemmBlocks, 256, 0, stream>>>(bufA, W, bufGW);
  walk_step  <<<walkBlocks, 256, 0, stream>>>(bufA, bufGW, mask, bufB);
  // iteration 2
  gemm_gW_wmma<<<gemmBlocks, 256, 0, stream>>>(bufB, W, bufGW);
  walk_step  <<<walkBlocks, 256, 0, stream>>>(bufB, bufGW, mask, out);
}